// EdgeEmbedding_35141422416182
// MI455X (gfx1250) — compile-verified
//
#include <hip/hip_runtime.h>

typedef __attribute__((ext_vector_type(2))) float v2f;
typedef __attribute__((ext_vector_type(8))) float v8f;

#define WAVES 4
#define BLOCK (WAVES * 32)
#define XSTRIDE 148                 // 144 payload + 4 pad: 16B-aligned rows, conflict-free A reads
#define WP_FLOATS (72 * 128)        // 144x64 W, pair-interleaved {W[2kp][c], W[2kp+1][c]}
#define X_FLOATS (WAVES * 16 * XSTRIDE)

__global__ __launch_bounds__(BLOCK)
void edge_embed_wmma_f32(const float* __restrict__ h,
                         const float* __restrict__ m,
                         const int*   __restrict__ eidx,   // (2, E) row-major
                         const float* __restrict__ W,      // (144, 64) row-major
                         float* __restrict__ out,          // (E, 64)
                         int nEdges) {
    __shared__ float Wp[WP_FLOATS];
    __shared__ float X[X_FLOATS];

    const int tid  = threadIdx.x;
    const int lane = tid & 31;
    const int wave = tid >> 5;
    const int n    = lane & 15;   // N / column / row-within-tile index
    const int hh   = lane >> 4;   // lane half

    // ---- Stage W into pair-interleaved LDS (coalesced float4 global reads) ----
    for (int i = tid; i < (144 * 64) / 4; i += BLOCK) {
        float4 w4 = ((const float4*)W)[i];
        int k  = i >> 4;          // row of W (16 float4 per row)
        int c0 = (i & 15) * 4;    // starting column
        float* dst = Wp + (k >> 1) * 128 + (k & 1);
        dst[(c0 + 0) * 2] = w4.x;
        dst[(c0 + 1) * 2] = w4.y;
        dst[(c0 + 2) * 2] = w4.z;
        dst[(c0 + 3) * 2] = w4.w;
    }

    // ---- Stage this wave's 16 edge rows: [h_src(64) | h_tgt(64) | m(16)] ----
    const int e0 = (blockIdx.x * WAVES + wave) * 16;
    float* Xw = X + wave * 16 * XSTRIDE;

    // h_src / h_tgt: 2 edges per iter, 16 lanes x float4 each (fully coalesced per row)
    #pragma unroll
    for (int i = 0; i < 8; ++i) {
        int r = i * 2 + hh;
        int e = e0 + r; if (e >= nEdges) e = nEdges - 1;   // clamp (stores are guarded)
        int s = eidx[e];
        int t = eidx[nEdges + e];
        float4 hs = ((const float4*)(h + (long long)s * 64))[n];
        float4 ht = ((const float4*)(h + (long long)t * 64))[n];
        ((float4*)(Xw + r * XSTRIDE))[n]      = hs;
        ((float4*)(Xw + r * XSTRIDE + 64))[n] = ht;
    }
    // m: 16 edges x 16 floats = 64 float4, coalesced
    #pragma unroll
    for (int i = 0; i < 2; ++i) {
        int t = i * 32 + lane;    // 0..63
        int r = t >> 2, q = t & 3;
        int e = e0 + r; if (e >= nEdges) e = nEdges - 1;
        float4 mv = ((const float4*)(m + (long long)e * 16))[q];
        ((float4*)(Xw + r * XSTRIDE + 128))[q] = mv;
    }

    __syncthreads();

    // ---- 16(edges) x 64(out) tile via V_WMMA_F32_16X16X4_F32, K = 144 = 36 x 4 ----
    v8f acc0 = {}, acc1 = {}, acc2 = {}, acc3 = {};
    #pragma unroll
    for (int kk = 0; kk < 36; ++kk) {
        // A fragment: lane holds row M = n, K = kk*4 + 2*hh + {0,1}
        v2f a = *(const v2f*)(Xw + n * XSTRIDE + kk * 4 + 2 * hh);
        // B fragments: lane holds col (nt*16 + n), K = kk*4 + 2*hh + {0,1}
        const float* wb = Wp + (kk * 2 + hh) * 128 + n * 2;
        v2f b0 = *(const v2f*)(wb + 0 * 32);
        v2f b1 = *(const v2f*)(wb + 1 * 32);
        v2f b2 = *(const v2f*)(wb + 2 * 32);
        v2f b3 = *(const v2f*)(wb + 3 * 32);
        acc0 = __builtin_amdgcn_wmma_f32_16x16x4_f32(false, a, false, b0, (short)0, acc0, false, false);
        acc1 = __builtin_amdgcn_wmma_f32_16x16x4_f32(false, a, false, b1, (short)0, acc1, false, false);
        acc2 = __builtin_amdgcn_wmma_f32_16x16x4_f32(false, a, false, b2, (short)0, acc2, false, false);
        acc3 = __builtin_amdgcn_wmma_f32_16x16x4_f32(false, a, false, b3, (short)0, acc3, false, false);
    }

    // ---- ScaledSiLU epilogue + store (two contiguous 64B segments per store op) ----
    const float SCALE = 1.0f / 0.6f;
    #pragma unroll
    for (int v = 0; v < 8; ++v) {
        int e = e0 + v + 8 * hh;          // C/D layout: lanes 0-15 -> M=v, 16-31 -> M=v+8
        if (e < nEdges) {
            float* o = out + (long long)e * 64 + n;
            float y;
            y = acc0[v]; o[0]  = SCALE * y / (1.0f + __expf(-y));
            y = acc1[v]; o[16] = SCALE * y / (1.0f + __expf(-y));
            y = acc2[v]; o[32] = SCALE * y / (1.0f + __expf(-y));
            y = acc3[v]; o[48] = SCALE * y / (1.0f + __expf(-y));
        }
    }
}

extern "C" void kernel_launch(void* const* d_in, const int* in_sizes, int n_in,
                              void* d_out, int out_size, void* d_ws, size_t ws_size,
                              hipStream_t stream) {
    const float* h    = (const float*)d_in[0];
    const float* m    = (const float*)d_in[1];
    const int*   eidx = (const int*)d_in[2];
    const float* W    = (const float*)d_in[3];
    float* out = (float*)d_out;

    int nEdges = in_sizes[1] / 16;
    int tiles  = (nEdges + 15) / 16;
    int blocks = (tiles + WAVES - 1) / WAVES;
    edge_embed_wmma_f32<<<blocks, BLOCK, 0, stream>>>(h, m, eidx, W, out, nEdges);
}